// DynamicASAPool_69200513074021
// MI455X (gfx1250) — compile-verified
//
#include <hip/hip_runtime.h>
#include <stdint.h>

#define FEAT   128
#define NGRAPH 32
#define GSZ    2048
#define NTOT   (NGRAPH*GSZ)      /* 65536 */
#define KNN    10
#define NEDGE  (NTOT*KNN)        /* 655360 */
#define KEEP   1024
#define NEG_SLOPE 0.2f
#define CCH    64                /* column-chunk (nodes) staged in LDS for kNN */

typedef float v2f __attribute__((ext_vector_type(2)));
typedef float v8f __attribute__((ext_vector_type(8)));

// ---------------- CDNA5 helpers ----------------
__device__ __forceinline__ uint32_t lds_off(const void* p) {
  // generic (flat) shared pointer: low 32 bits are the LDS offset
  return (uint32_t)(uintptr_t)p;
}
__device__ __forceinline__ void async_copy16(uint32_t lds, const void* g) {
  asm volatile("global_load_async_to_lds_b128 %0, %1, off"
               :: "v"(lds), "v"(g) : "memory");
}
__device__ __forceinline__ void wait_async() {
  asm volatile("s_wait_asynccnt 0" ::: "memory");
}
__device__ __forceinline__ void wait_ds() {
  asm volatile("s_wait_dscnt 0" ::: "memory");
}
__device__ __forceinline__ float wsum(float v) {
#pragma unroll
  for (int m = 16; m; m >>= 1) v += __shfl_xor(v, m, 32);
  return v;
}

// ---------------- 1) row squared norms ----------------
__global__ __launch_bounds__(256) void k_sq(const float* __restrict__ x,
                                            float* __restrict__ sq) {
  int node = blockIdx.x * 8 + (threadIdx.x >> 5);
  int lane = threadIdx.x & 31;
  float4 v = ((const float4*)(x + (size_t)node * FEAT))[lane];
  float s = v.x * v.x + v.y * v.y + v.z * v.z + v.w * v.w;
  s = wsum(s);
  if (lane == 0) sq[node] = s;
}

// ---------------- 2) fused WMMA distance + per-row top-K ----------------
// grid = NGRAPH*16 blocks, 256 threads (8 waves); wave owns a 16-row tile.
// Two independent C accumulators (even/odd K chunks) break the WMMA RAW
// chain so the XDL pipe can overlap successive v_wmma issues.
__global__ __launch_bounds__(256) void k_knn(const float* __restrict__ x,
                                             const float* __restrict__ sq,
                                             int* __restrict__ knn) {
  __shared__ float sx[CCH][FEAT + 4];   // 64 x 132 f32 = 33 KB (pad vs banks)
  __shared__ float ssq[CCH];
  __shared__ float tile[8][16 * 17];    // per-wave 16x16 score tile, pitch 17

  const int g   = blockIdx.x >> 4;
  const int rb  = (blockIdx.x & 15) << 7;   // 128 rows per block
  const int wav = threadIdx.x >> 5;
  const int lane = threadIdx.x & 31;
  const int l16 = lane & 15;
  const int hf  = lane >> 4;

  const float* xg  = x  + (size_t)g * GSZ * FEAT;
  const float* sqg = sq + g * GSZ;
  const int row = rb + wav * 16 + l16;

  // A panel in WMMA 16x4 A-layout: lane m=l16, K pair = kc*4 + hf*2
  v2f A[32];
#pragma unroll
  for (int kc = 0; kc < 32; ++kc) {
    int k0 = kc * 4 + hf * 2;
    A[kc] = *(const v2f*)(xg + (size_t)row * FEAT + k0);
  }

  float best[KNN]; int bidx[KNN];
#pragma unroll
  for (int t = 0; t < KNN; ++t) { best[t] = -3.4e38f; bidx[t] = 0; }

  for (int cb = 0; cb < GSZ; cb += CCH) {
    __syncthreads();                                   // prior reads done
    for (int u = threadIdx.x; u < CCH * 32; u += 256) { // 16B units
      int r = u >> 5, c = u & 31;
      async_copy16(lds_off(&sx[r][c * 4]),
                   xg + (size_t)(cb + r) * FEAT + c * 4);
    }
    if (threadIdx.x < CCH / 4)
      async_copy16(lds_off(&ssq[threadIdx.x * 4]), sqg + cb + threadIdx.x * 4);
    wait_async();
    __syncthreads();

    for (int ct = 0; ct < CCH / 16; ++ct) {
      int nl = ct * 16 + l16;            // this lane's B column (N index)
      v8f c0, c1;
#pragma unroll
      for (int i = 0; i < 8; ++i) { c0[i] = 0.f; c1[i] = 0.f; }
#pragma unroll
      for (int kc = 0; kc < 16; ++kc) {
        int ka = (2 * kc) * 4 + hf * 2;
        int kb = (2 * kc + 1) * 4 + hf * 2;
        v2f ba = *(const v2f*)&sx[nl][ka];
        v2f bb = *(const v2f*)&sx[nl][kb];
        c0 = __builtin_amdgcn_wmma_f32_16x16x4_f32(false, A[2 * kc], false, ba,
                                                   (short)0, c0, false, false);
        c1 = __builtin_amdgcn_wmma_f32_16x16x4_f32(false, A[2 * kc + 1], false, bb,
                                                   (short)0, c1, false, false);
      }
      float sqj = ssq[nl];
      // candidate score s = 2*dot - sq_j  (sq_i row-constant: rank-invariant)
#pragma unroll
      for (int i = 0; i < 8; ++i)
        tile[wav][l16 * 17 + hf * 8 + i] = 2.f * (c0[i] + c1[i]) - sqj;
      wait_ds();
      // per-row top-K insertion; lanes 16-31 duplicate row l16 (discarded)
#pragma unroll
      for (int n = 0; n < 16; ++n) {
        float s = tile[wav][n * 17 + l16];
        int j = cb + ct * 16 + n;
        if (s > best[KNN - 1]) {
          best[KNN - 1] = s; bidx[KNN - 1] = j;
#pragma unroll
          for (int t = KNN - 1; t > 0; --t) {
            if (best[t] > best[t - 1]) {
              float tv = best[t]; best[t] = best[t - 1]; best[t - 1] = tv;
              int ti = bidx[t];  bidx[t] = bidx[t - 1];  bidx[t - 1] = ti;
            }
          }
        }
      }
    }
  }
  if (lane < 16) {
    int grow = g * GSZ + row;
#pragma unroll
    for (int t = 0; t < KNN; ++t)
      knn[(size_t)grow * KNN + t] = g * GSZ + bidx[t];
  }
}

// ---------------- 3) reverse-CSR build ----------------
__global__ void k_zero_i(int* p, int n) {
  int i = blockIdx.x * 256 + threadIdx.x;
  if (i < n) p[i] = 0;
}
__global__ void k_deg(const int* __restrict__ knn, int* __restrict__ deg) {
  int e = blockIdx.x * 256 + threadIdx.x;
  if (e < NEDGE) atomicAdd(&deg[knn[e]], 1);
}
__global__ __launch_bounds__(1024) void k_scan(const int* __restrict__ deg,
                                               int* __restrict__ rs,
                                               int* __restrict__ cur) {
  __shared__ int buf[1024];
  __shared__ int carry;
  int tid = threadIdx.x;
  if (tid == 0) carry = 0;
  __syncthreads();
  for (int ch = 0; ch < NTOT / 1024; ++ch) {
    int v = deg[ch * 1024 + tid];
    buf[tid] = v; __syncthreads();
    for (int off = 1; off < 1024; off <<= 1) {
      int t = (tid >= off) ? buf[tid - off] : 0;
      __syncthreads();
      buf[tid] += t;
      __syncthreads();
    }
    int ex = carry + buf[tid] - v;   // exclusive
    rs[ch * 1024 + tid] = ex; cur[ch * 1024 + tid] = ex;
    __syncthreads();
    if (tid == 0) carry += buf[1023];
    __syncthreads();
  }
  if (tid == 0) rs[NTOT] = NEDGE;
}
__global__ void k_fill(const int* __restrict__ knn, int* __restrict__ cur,
                       int* __restrict__ csr) {
  int e = blockIdx.x * 256 + threadIdx.x;
  if (e < NEDGE) {
    int dst = knn[e];
    int pos = atomicAdd(&cur[dst], 1);
    csr[pos] = e / KNN;              // src (query) node of this edge
  }
}

// ---------------- 4) x_q = segment_max(x[src], dst) via gather ----------------
__global__ __launch_bounds__(256) void k_xqmax(const float* __restrict__ x,
                                               const int* __restrict__ rs,
                                               const int* __restrict__ csr,
                                               float* __restrict__ xq) {
  int node = blockIdx.x * 8 + (threadIdx.x >> 5);
  int lane = threadIdx.x & 31;
  int beg = rs[node], end = rs[node + 1];
  float4 acc = make_float4(-3.4e38f, -3.4e38f, -3.4e38f, -3.4e38f);
  for (int t = beg; t < end; ++t) {
    float4 v = ((const float4*)(x + (size_t)csr[t] * FEAT))[lane];
    acc.x = fmaxf(acc.x, v.x); acc.y = fmaxf(acc.y, v.y);
    acc.z = fmaxf(acc.z, v.z); acc.w = fmaxf(acc.w, v.w);
  }
  ((float4*)(xq + (size_t)node * FEAT))[lane] = acc;
}

// ---------------- 5) x_q @ lin_W + lin_b via WMMA ----------------
__global__ __launch_bounds__(256) void k_lin(const float* __restrict__ xq,
                                             const float* __restrict__ W,
                                             const float* __restrict__ bias,
                                             float* __restrict__ out) {
  const int wav = threadIdx.x >> 5, lane = threadIdx.x & 31;
  const int l16 = lane & 15, hf = lane >> 4;
  const size_t rowbase = (size_t)blockIdx.x * 128 + wav * 16;
  const int row = (int)rowbase + l16;
  v2f A[32];
#pragma unroll
  for (int kc = 0; kc < 32; ++kc) {
    int k0 = kc * 4 + hf * 2;
    A[kc] = *(const v2f*)(xq + (size_t)row * FEAT + k0);
  }
  for (int nt = 0; nt < 8; ++nt) {
    int n = nt * 16 + l16;
    float bv = bias[n];
    v8f c0, c1;
#pragma unroll
    for (int i = 0; i < 8; ++i) { c0[i] = bv; c1[i] = 0.f; }
#pragma unroll
    for (int kc = 0; kc < 16; ++kc) {
      int ka = (2 * kc) * 4 + hf * 2;
      int kb = (2 * kc + 1) * 4 + hf * 2;
      v2f ba, bb;
      ba.x = W[(size_t)ka * FEAT + n];
      ba.y = W[(size_t)(ka + 1) * FEAT + n];
      bb.x = W[(size_t)kb * FEAT + n];
      bb.y = W[(size_t)(kb + 1) * FEAT + n];
      c0 = __builtin_amdgcn_wmma_f32_16x16x4_f32(false, A[2 * kc], false, ba,
                                                 (short)0, c0, false, false);
      c1 = __builtin_amdgcn_wmma_f32_16x16x4_f32(false, A[2 * kc + 1], false, bb,
                                                 (short)0, c1, false, false);
    }
#pragma unroll
    for (int i = 0; i < 8; ++i)
      out[(rowbase + hf * 8 + i) * FEAT + n] = c0[i] + c1[i];
  }
}

// ---------------- 6) per-node attention dots ----------------
__global__ __launch_bounds__(256) void k_pq(const float* __restrict__ x,
                                            const float* __restrict__ xql,
                                            const float* __restrict__ attW,
                                            float* __restrict__ p,
                                            float* __restrict__ q) {
  int node = blockIdx.x * 8 + (threadIdx.x >> 5);
  int lane = threadIdx.x & 31;
  float4 w1 = ((const float4*)attW)[lane];            // att_W[:F]
  float4 w2 = ((const float4*)(attW + FEAT))[lane];   // att_W[F:]
  float4 xv = ((const float4*)(x + (size_t)node * FEAT))[lane];
  float4 qv = ((const float4*)(xql + (size_t)node * FEAT))[lane];
  float ps = xv.x * w2.x + xv.y * w2.y + xv.z * w2.z + xv.w * w2.w;
  float qs = qv.x * w1.x + qv.y * w1.y + qv.z * w1.z + qv.w * w1.w;
  ps = wsum(ps); qs = wsum(qs);
  if (lane == 0) { p[node] = ps; q[node] = qs; }
}

// ---------------- 7) segment softmax + weighted sum -> x_new ----------------
__global__ __launch_bounds__(256) void k_xnew(const float* __restrict__ x,
                                              const int* __restrict__ rs,
                                              const int* __restrict__ csr,
                                              const float* __restrict__ p,
                                              const float* __restrict__ q,
                                              const float* __restrict__ attb,
                                              float* __restrict__ xnew) {
  int node = blockIdx.x * 8 + (threadIdx.x >> 5);
  int lane = threadIdx.x & 31;
  int beg = rs[node], end = rs[node + 1];
  float qv = q[node], ab = attb[0];
  // online (m, d) softmax stats per lane, then cross-lane merge
  float m = -3.4e38f, d = 0.f;
  for (int t = beg + lane; t < end; t += 32) {
    float sc = qv + p[csr[t]] + ab;
    sc = sc > 0.f ? sc : NEG_SLOPE * sc;
    float nm = fmaxf(m, sc);
    d = d * __expf(m - nm) + __expf(sc - nm);
    m = nm;
  }
#pragma unroll
  for (int off = 16; off; off >>= 1) {
    float m2 = __shfl_xor(m, off, 32);
    float d2 = __shfl_xor(d, off, 32);
    float nm = fmaxf(m, m2);
    d = d * __expf(m - nm) + d2 * __expf(m2 - nm);
    m = nm;
  }
  float inv = 1.f / (d + 1e-16f);
  float4 acc = make_float4(0.f, 0.f, 0.f, 0.f);
  for (int t = beg; t < end; ++t) {
    int s = csr[t];
    float sc = qv + p[s] + ab;
    sc = sc > 0.f ? sc : NEG_SLOPE * sc;
    float w = __expf(sc - m) * inv;
    float4 xv = ((const float4*)(x + (size_t)s * FEAT))[lane];
    acc.x += w * xv.x; acc.y += w * xv.y; acc.z += w * xv.z; acc.w += w * xv.w;
  }
  ((float4*)(xnew + (size_t)node * FEAT))[lane] = acc;
}

// ---------------- 8) LEConv per-node projections ----------------
__global__ __launch_bounds__(256) void k_abc(const float* __restrict__ xnew,
                                             const float* __restrict__ w1,
                                             const float* __restrict__ b1,
                                             const float* __restrict__ w2,
                                             const float* __restrict__ w3,
                                             float* __restrict__ a,
                                             float* __restrict__ b,
                                             float* __restrict__ c3) {
  int node = blockIdx.x * 8 + (threadIdx.x >> 5);
  int lane = threadIdx.x & 31;
  float4 xv = ((const float4*)(xnew + (size_t)node * FEAT))[lane];
  float4 v1 = ((const float4*)w1)[lane];
  float4 v2 = ((const float4*)w2)[lane];
  float4 v3 = ((const float4*)w3)[lane];
  float s1 = xv.x * v1.x + xv.y * v1.y + xv.z * v1.z + xv.w * v1.w;
  float s2 = xv.x * v2.x + xv.y * v2.y + xv.z * v2.z + xv.w * v2.w;
  float s3 = xv.x * v3.x + xv.y * v3.y + xv.z * v3.z + xv.w * v3.w;
  s1 = wsum(s1); s2 = wsum(s2); s3 = wsum(s3);
  if (lane == 0) { a[node] = s1 + b1[0]; b[node] = s2; c3[node] = s3; }
}

// ---------------- 9) fitness ----------------
__global__ __launch_bounds__(256) void k_fit(const int* __restrict__ rs,
                                             const int* __restrict__ csr,
                                             const float* __restrict__ a,
                                             const float* __restrict__ b,
                                             const float* __restrict__ c3,
                                             const float* __restrict__ b3,
                                             float* __restrict__ fit) {
  int j = blockIdx.x * 256 + threadIdx.x;
  if (j >= NTOT) return;
  int beg = rs[j], end = rs[j + 1];
  float s = 0.f;
  for (int t = beg; t < end; ++t) s += a[csr[t]];
  float v = s - (float)(end - beg) * b[j] + c3[j] + b3[0];
  fit[j] = 1.f / (1.f + __expf(-v));
}

// ---------------- 10) per-graph top-1024 (bitonic) + output ----------------
__global__ __launch_bounds__(1024) void k_sel(const float* __restrict__ fit,
                                              const float* __restrict__ xnew,
                                              float* __restrict__ outF,
                                              int* __restrict__ outB) {
  __shared__ float v[GSZ];
  __shared__ int id[GSZ];
  int g = blockIdx.x, tid = threadIdx.x;
  for (int i = tid; i < GSZ; i += 1024) { v[i] = fit[g * GSZ + i]; id[i] = i; }
  __syncthreads();
  for (int k = 2; k <= GSZ; k <<= 1) {
    for (int jj = k >> 1; jj > 0; jj >>= 1) {
      for (int i = tid; i < GSZ; i += 1024) {
        int ix = i ^ jj;
        if (ix > i) {
          bool desc = ((i & k) == 0);
          bool bad = desc ? (v[i] < v[ix] || (v[i] == v[ix] && id[i] > id[ix]))
                          : (v[i] > v[ix] || (v[i] == v[ix] && id[i] < id[ix]));
          if (bad) {
            float tv = v[i]; v[i] = v[ix]; v[ix] = tv;
            int ti = id[i]; id[i] = id[ix]; id[ix] = ti;
          }
        }
      }
      __syncthreads();
    }
  }
  const float4* xn4 = (const float4*)xnew;
  float4* o4 = (float4*)outF;
  for (int u = tid; u < KEEP * 32; u += 1024) {
    int r = u >> 5, c = u & 31;
    int sel = id[r]; float fv = v[r];
    float4 xv = xn4[(size_t)(g * GSZ + sel) * 32 + c];
    xv.x *= fv; xv.y *= fv; xv.z *= fv; xv.w *= fv;
    o4[(size_t)(g * KEEP + r) * 32 + c] = xv;
  }
  for (int r = tid; r < KEEP; r += 1024) outB[g * KEEP + r] = g;
}

// ---------------- host launcher ----------------
extern "C" void kernel_launch(void* const* d_in, const int* in_sizes, int n_in,
                              void* d_out, int out_size, void* d_ws, size_t ws_size,
                              hipStream_t stream) {
  (void)in_sizes; (void)n_in; (void)out_size; (void)ws_size;
  const float* x    = (const float*)d_in[0];
  const float* linW = (const float*)d_in[2];
  const float* linb = (const float*)d_in[3];
  const float* attW = (const float*)d_in[4];
  const float* attb = (const float*)d_in[5];
  const float* le1W = (const float*)d_in[6];
  const float* le1b = (const float*)d_in[7];
  const float* le2W = (const float*)d_in[8];
  const float* le3W = (const float*)d_in[9];
  const float* le3b = (const float*)d_in[10];

  char* ws = (char*)d_ws;
  size_t off = 0;
  auto alloc = [&](size_t bytes) -> char* {
    char* p = ws + off; off += (bytes + 255) & ~(size_t)255; return p;
  };
  float* sq   = (float*)alloc((size_t)NTOT * 4);
  int*   knn  = (int*)  alloc((size_t)NEDGE * 4);
  int*   deg  = (int*)  alloc((size_t)NTOT * 4);
  int*   rs   = (int*)  alloc((size_t)(NTOT + 1) * 4);
  int*   cur  = (int*)  alloc((size_t)NTOT * 4);
  int*   csr  = (int*)  alloc((size_t)NEDGE * 4);
  float* xq   = (float*)alloc((size_t)NTOT * FEAT * 4);
  float* xql  = (float*)alloc((size_t)NTOT * FEAT * 4);
  float* pbuf = (float*)alloc((size_t)NTOT * 4);
  float* qbuf = (float*)alloc((size_t)NTOT * 4);
  float* xnew = (float*)alloc((size_t)NTOT * FEAT * 4);
  float* abuf = (float*)alloc((size_t)NTOT * 4);
  float* bbuf = (float*)alloc((size_t)NTOT * 4);
  float* cbuf = (float*)alloc((size_t)NTOT * 4);
  float* fit  = (float*)alloc((size_t)NTOT * 4);

  float* outF = (float*)d_out;
  int*   outB = (int*)(outF + (size_t)NGRAPH * KEEP * FEAT);

  k_sq  <<<NTOT / 8,            256,  0, stream>>>(x, sq);
  k_knn <<<NGRAPH * 16,         256,  0, stream>>>(x, sq, knn);
  k_zero_i<<<NTOT / 256,        256,  0, stream>>>(deg, NTOT);
  k_deg <<<(NEDGE + 255) / 256, 256,  0, stream>>>(knn, deg);
  k_scan<<<1,                   1024, 0, stream>>>(deg, rs, cur);
  k_fill<<<(NEDGE + 255) / 256, 256,  0, stream>>>(knn, cur, csr);
  k_xqmax<<<NTOT / 8,           256,  0, stream>>>(x, rs, csr, xq);
  k_lin <<<NTOT / 128,          256,  0, stream>>>(xq, linW, linb, xql);
  k_pq  <<<NTOT / 8,            256,  0, stream>>>(x, xql, attW, pbuf, qbuf);
  k_xnew<<<NTOT / 8,            256,  0, stream>>>(x, rs, csr, pbuf, qbuf, attb, xnew);
  k_abc <<<NTOT / 8,            256,  0, stream>>>(xnew, le1W, le1b, le2W, le3W, abuf, bbuf, cbuf);
  k_fit <<<NTOT / 256,          256,  0, stream>>>(rs, csr, abuf, bbuf, cbuf, le3b, fit);
  k_sel <<<NGRAPH,              1024, 0, stream>>>(fit, xnew, outF, outB);
}